// X_debaising_37151467111219
// MI455X (gfx1250) — compile-verified
//
#include <hip/hip_runtime.h>

// out[i,j] = feature[i,j] * s[j]   (N=500000 rows, F=256 cols, fp32)
// Memory-bound streamer: 1.02 GB total traffic, ~44 us floor at 23.3 TB/s.
// WMMA is intentionally absent: arithmetic intensity is 0.125 FLOP/byte, so
// performance is set entirely by the b128 + NT-hint vector memory path.

typedef float v4f __attribute__((ext_vector_type(4)));

// Fast path: F == 256 -> 64 float4 column-groups per row.
// Block = 256 threads (8 wave32) = 4 rows x 64 column-groups. Lanes 0..31 of
// each wave share one row (threadIdx.x>>6 is wave-uniform), so we pin the row
// index into an SGPR via readfirstlane: all loop control is scalar
// (s_cmp/s_cbranch), and the per-lane work is exactly one global_load_b128(NT),
// four packed multiplies, one global_store_b128(NT) per iteration.
__global__ void scale_cols_f256_kernel(const float* __restrict__ feature,
                                       const float* __restrict__ s,
                                       float* __restrict__ out,
                                       int nrows) {
    const int colg = threadIdx.x & 63;                 // per-lane: float4 col group
    // Wave-uniform by construction; readfirstlane makes that visible to the
    // compiler so loop control stays in SALU instead of v_cmpx/exec masking.
    const int rsub = __builtin_amdgcn_readfirstlane(threadIdx.x >> 6);

    // s is 1 KB total; one 16B load per thread, then register-resident.
    const v4f sv = ((const v4f*)s)[colg];

    const v4f* __restrict__ fin  = (const v4f*)feature;
    v4f*       __restrict__ fout = (v4f*)out;

    const int stride = gridDim.x * 4;                  // rows advanced per sweep
    int row = blockIdx.x * 4 + rsub;                   // scalar

    #pragma unroll 4
    for (; row < nrows; row += stride) {
        const size_t idx = (size_t)row * 64 + (size_t)colg;

        // Unconditional speculative prefetch of the next sweep (TH=0 per
        // ISA 7.3: translation failures silently dropped, so running at most
        // one stride past the buffer end is safe). Emits global_prefetch_b8.
        __builtin_prefetch((const void*)&fin[idx + (size_t)stride * 64], 0, 0);

        // NT temporal hint: every byte is touched exactly once and the 1 GB
        // working set is >> 192 MB L2, so stream around the cache.
        v4f v = __builtin_nontemporal_load(&fin[idx]);
        v *= sv;
        __builtin_nontemporal_store(v, &fout[idx]);
    }
}

// Generic fallback (any F % 4 == 0): flat float4 grid-stride, 64-bit modulo
// for the column group; s stays hot in cache. Unused for reference shapes.
__global__ void scale_cols_v4_generic_kernel(const float* __restrict__ feature,
                                             const float* __restrict__ s,
                                             float* __restrict__ out,
                                             unsigned long long total4,
                                             unsigned colGroups) {
    const v4f* __restrict__ fin  = (const v4f*)feature;
    const v4f* __restrict__ sv4  = (const v4f*)s;
    v4f*       __restrict__ fout = (v4f*)out;

    unsigned long long i = (unsigned long long)blockIdx.x * blockDim.x + threadIdx.x;
    const unsigned long long gstride = (unsigned long long)gridDim.x * blockDim.x;
    for (; i < total4; i += gstride) {
        const unsigned cg = (unsigned)(i % colGroups);
        v4f v = __builtin_nontemporal_load(&fin[i]);
        v *= sv4[cg];
        __builtin_nontemporal_store(v, &fout[i]);
    }
}

// Last-resort scalar path (odd F). Unused for reference shapes.
__global__ void scale_cols_scalar_kernel(const float* __restrict__ feature,
                                         const float* __restrict__ s,
                                         float* __restrict__ out,
                                         unsigned long long total, unsigned F) {
    unsigned long long i = (unsigned long long)blockIdx.x * blockDim.x + threadIdx.x;
    const unsigned long long gstride = (unsigned long long)gridDim.x * blockDim.x;
    for (; i < total; i += gstride) {
        out[i] = feature[i] * s[(unsigned)(i % F)];
    }
}

extern "C" void kernel_launch(void* const* d_in, const int* in_sizes, int n_in,
                              void* d_out, int out_size, void* d_ws, size_t ws_size,
                              hipStream_t stream) {
    const float* feature = (const float*)d_in[0];
    const float* s       = (const float*)d_in[1];
    float*       out     = (float*)d_out;

    const int F = in_sizes[1];                       // 256 in the reference
    const unsigned long long total = (unsigned long long)in_sizes[0];
    const int nrows = (int)(total / (unsigned long long)F);

    if (F == 256) {
        // 8192 blocks x 8 wave32 = 64K waves: maximal MLP for the HBM
        // streamer; each thread still runs ~15 grid-stride iterations.
        int blocks = 8192;
        const int maxBlocks = (nrows + 3) / 4;
        if (blocks > maxBlocks) blocks = maxBlocks;
        if (blocks < 1) blocks = 1;
        scale_cols_f256_kernel<<<blocks, 256, 0, stream>>>(feature, s, out, nrows);
    } else if ((F & 3) == 0) {
        scale_cols_v4_generic_kernel<<<8192, 256, 0, stream>>>(
            feature, s, out, total / 4ull, (unsigned)(F / 4));
    } else {
        scale_cols_scalar_kernel<<<8192, 256, 0, stream>>>(
            feature, s, out, total, (unsigned)F);
    }
}